// GLinear_1116691497444
// MI455X (gfx1250) — compile-verified
//
#include <hip/hip_runtime.h>

typedef float v2f __attribute__((ext_vector_type(2)));
typedef float v4f __attribute__((ext_vector_type(4)));
typedef float v8f __attribute__((ext_vector_type(8)));

#define AS1 __attribute__((address_space(1)))
#define AS3 __attribute__((address_space(3)))

#define C_DIM  256
#define D_DIM  16
#define KCHUNK 32
#define PADK   36     // within-plane row stride (floats): conflict-free b64 A-frag reads
#define PLANE  578    // d-plane stride (floats); 578 % 64 == 2 -> the 16 consecutive-d
                      // lanes of an async staging wave hit 16 distinct banks

#if defined(__gfx1250__) && __has_builtin(__builtin_amdgcn_global_load_async_to_lds_b32)
#define USE_ASYNC_STAGE 1
#else
#define USE_ASYNC_STAGE 0
#endif

__global__ __launch_bounds__(512, 1) void glinear_wmma_f32(
    const float* __restrict__ x, const float* __restrict__ W00,
    const float* __restrict__ W10, const float* __restrict__ W11,
    float* __restrict__ out)
{
    // x chunk staged transposed: xs[d][p][k], padded plane + row strides
    __shared__ __attribute__((aligned(16))) float xs[(D_DIM - 1) * PLANE + 16 * PADK];

    const int t    = threadIdx.x;
    const int lane = t & 31;
    const int wv   = t >> 5;        // 16 waves -> one 16-wide N-tile each
    const int lm   = lane & 15;     // M for A-frag rows, N for B/C-frag cols
    const int lh   = lane >> 4;     // half-wave selector
    const int p0   = blockIdx.x * 16;
    const int n0   = wv * 16;

    v8f acc[D_DIM] = {};            // 16 d-channels x 8 VGPRs accumulators

    for (int kc = 0; kc < C_DIM; kc += KCHUNK) {
        __syncthreads();
#if USE_ASYNC_STAGE
        {
            // Each thread owns one (d, i) column; async engine writes LDS directly.
            // Global side: 32 lanes cover 128 contiguous bytes. LDS side: bank-free.
            const int d = t & 15;
            const int i = t >> 4;   // 0..31
            const float* gsrc = x + ((size_t)p0 * C_DIM + (kc + i)) * D_DIM + d;
            float* ldst = &xs[d * PLANE + i];
            #pragma unroll
            for (int p = 0; p < 16; ++p) {
                __builtin_amdgcn_global_load_async_to_lds_b32(
                    (AS1 int*)(gsrc + (size_t)p * C_DIM * D_DIM),
                    (AS3 int*)(ldst + p * PADK),
                    0, 0);
            }
#if __has_builtin(__builtin_amdgcn_s_wait_asynccnt)
            __builtin_amdgcn_s_wait_asynccnt(0);
#else
            asm volatile("s_wait_asynccnt 0x0" ::: "memory");
#endif
        }
#else
        // Fallback: coalesced float4 loads + LDS scatter-transpose through VGPRs.
        #pragma unroll
        for (int it = 0; it < 4; ++it) {
            const int idx = it * 512 + t;
            const int d4  = idx & 3;
            const int ci  = (idx >> 2) & (KCHUNK - 1);
            const int p   = idx >> 7;
            const float4 v =
                ((const float4*)x)[((size_t)(p0 + p) * C_DIM + (kc + ci)) * 4 + d4];
            float* dst = &xs[(d4 * 4) * PLANE + p * PADK + ci];
            dst[0 * PLANE] = v.x;
            dst[1 * PLANE] = v.y;
            dst[2 * PLANE] = v.z;
            dst[3 * PLANE] = v.w;
        }
#endif
        __syncthreads();

        #pragma unroll
        for (int ks = 0; ks < KCHUNK; ks += 4) {
            const int kg = kc + ks + 2 * lh;         // global K col for B frag
            const int kl = ks + 2 * lh;              // LDS K col for A frag

            // ---- group W00 : d = 0
            {
                const v2f b = *(const v2f*)&W00[(size_t)(n0 + lm) * C_DIM + kg];
                const v2f a = *(const v2f*)&xs[0 * PLANE + lm * PADK + kl];
                acc[0] = __builtin_amdgcn_wmma_f32_16x16x4_f32(
                    false, a, false, b, (short)0, acc[0], false, false);
            }
            // ---- group W10 : d = 1..6 (one B fragment feeds 6 WMMAs)
            {
                const v2f b = *(const v2f*)&W10[(size_t)(n0 + lm) * C_DIM + kg];
                #pragma unroll
                for (int d = 1; d < 7; ++d) {
                    const v2f a = *(const v2f*)&xs[d * PLANE + lm * PADK + kl];
                    acc[d] = __builtin_amdgcn_wmma_f32_16x16x4_f32(
                        false, a, false, b, (short)0, acc[d], false, false);
                }
            }
            // ---- group W11 : d = 7..15 (one B fragment feeds 9 WMMAs)
            {
                const v2f b = *(const v2f*)&W11[(size_t)(n0 + lm) * C_DIM + kg];
                #pragma unroll
                for (int d = 7; d < 16; ++d) {
                    const v2f a = *(const v2f*)&xs[d * PLANE + lm * PADK + kl];
                    acc[d] = __builtin_amdgcn_wmma_f32_16x16x4_f32(
                        false, a, false, b, (short)0, acc[d], false, false);
                }
            }
        }
    }

    // Epilogue: residual add (+x, last use -> NT load) and dense 128-bit NT stores
    // over the d axis. C/D frag layout: VGPR r -> M = r + 8*lh, N = lm.
    #pragma unroll
    for (int r = 0; r < 8; ++r) {
        const size_t base =
            ((size_t)(p0 + r + 8 * lh) * C_DIM + (n0 + lm)) * D_DIM;
        const v4f* xr = (const v4f*)(x + base);
        v4f* orow     = (v4f*)(out + base);
        #pragma unroll
        for (int q = 0; q < 4; ++q) {
            v4f v = __builtin_nontemporal_load(xr + q);
            v[0] += acc[q * 4 + 0][r];
            v[1] += acc[q * 4 + 1][r];
            v[2] += acc[q * 4 + 2][r];
            v[3] += acc[q * 4 + 3][r];
            __builtin_nontemporal_store(v, orow + q);
        }
    }
}

extern "C" void kernel_launch(void* const* d_in, const int* in_sizes, int n_in,
                              void* d_out, int out_size, void* d_ws, size_t ws_size,
                              hipStream_t stream) {
    const float* x   = (const float*)d_in[0];
    const float* W00 = (const float*)d_in[1];
    const float* W10 = (const float*)d_in[2];
    const float* W11 = (const float*)d_in[3];
    float* out       = (float*)d_out;

    const int P = in_sizes[0] / (C_DIM * D_DIM);   // 8192
    glinear_wmma_f32<<<P / 16, 512, 0, stream>>>(x, W00, W10, W11, out);
}